// NV12ToRgb_5781025980441
// MI455X (gfx1250) — compile-verified
//
#include <hip/hip_runtime.h>

typedef float v2f __attribute__((ext_vector_type(2)));
typedef float v4f __attribute__((ext_vector_type(4)));
typedef float v8f __attribute__((ext_vector_type(8)));

namespace {
constexpr int kHalfH = 2160;
constexpr int kHalfW = 3840;
constexpr int kFullW = kHalfW * 2;                         // 7680
constexpr int kYPlane = kFullW * (kHalfH * 2);             // 33,177,600 dwords
constexpr int kTilesPerWave = 8;                           // 16 px per WMMA tile
constexpr int kWavesPerBlock = 8;                          // wave32 -> 256 threads
constexpr int kBlockThreads = kWavesPerBlock * 32;
constexpr int kPixPerWave = kTilesPerWave * 16;            // 128 (divides 3840)
constexpr int kPixPerBlock = kWavesPerBlock * kPixPerWave; // 1024
}  // namespace

__device__ __forceinline__ void wave_lds_fence() {
  // Per-wave LDS producer/consumer ordering: DS ops from one wave complete
  // in order, so a dscnt wait + compiler scheduling barrier is sufficient
  // (no workgroup barrier needed; each wave owns its LDS slice).
  __builtin_amdgcn_wave_barrier();
#if __has_builtin(__builtin_amdgcn_s_wait_dscnt)
  __builtin_amdgcn_s_wait_dscnt(0);
#else
  asm volatile("s_wait_dscnt 0" ::: "memory");
#endif
  __builtin_amdgcn_wave_barrier();
}

// D(16x16) = A(16x4) x B(4x16) + 0
//   M = output channel (rows 0..2 = R,G,B coeffs; K=3 col = folded bias)
//   N = pixel (16 consecutive output pixels per tile)
//   K = {Y, U, V, 1}
// f32 operand striping (per ISA 16x16x4 layout):
//   A/B V0: lanes0-15 = K0, lanes16-31 = K2 ; V1: K1 / K3
//   D  Vr: lanes0-15 = (M=r, N=lane) -> V0=R, V1=G, V2=B planar across lanes.
__global__ __launch_bounds__(kBlockThreads) void nv12_to_rgb_wmma(
    const int* __restrict__ bytes, const float* __restrict__ mat,
    const float* __restrict__ off, float* __restrict__ out) {
  __shared__ __align__(16) float lds[kWavesPerBlock * 64];

  const int tid = threadIdx.x;
  const int wave = tid >> 5;
  const int lane = tid & 31;
  const int half = lane >> 4;   // 0: K0/K1 operand half, 1: K2/K3 half
  const int l16 = lane & 15;

  // Coefficients (uniform pointers -> scalar loads). Fold YUV offset into a
  // per-channel bias so the K=3 "ones" row carries it:
  //   rgb[m] = sum_k M[k][m]*yuv[k] + bias[m],  bias[m] = -sum_k off[k]*M[k][m]
  const float o0 = off[0], o1 = off[1], o2 = off[2];
  float mk0 = 0.0f, mk1 = 0.0f, mk2 = 0.0f, bias = 0.0f;
  if (l16 < 3) {
    mk0 = mat[0 + l16];   // M[0][m]
    mk1 = mat[3 + l16];   // M[1][m]
    mk2 = mat[6 + l16];   // M[2][m]
    bias = -(o0 * mk0 + o1 * mk1 + o2 * mk2);
  }
  v2f A;
  A.x = half ? mk2 : mk0;   // V0: K0 (lanes0-15) | K2 (lanes16-31)
  A.y = half ? bias : mk1;  // V1: K1            | K3 (bias)

  // ---- All address math hoisted out of the tile loop ----
  // 3840 % 128 == 0, so each wave's 128-pixel span lies within ONE chroma row:
  // the row/col split (one integer division) is per-wave loop-invariant, and
  // per-tile address deltas are compile-time constants that fold into the
  // 24-bit vmem instruction offsets.
  const int pixBase = (int)blockIdx.x * kPixPerBlock + wave * kPixPerWave;
  const int i = pixBase / kHalfW;              // chroma row (wave-uniform)
  const int j0 = pixBase - i * kHalfW;         // chroma col of pixel 0

  const int yBase = (2 * i) * kFullW + 2 * (j0 + l16);  // even row, even col
  const int uvBase = kYPlane + 2 * (pixBase + l16);     // interleaved U,V
  const int* const p0 = bytes + (half ? (uvBase + 1) : yBase);  // Y | V stream
  const int* const pu = bytes + uvBase;                         // U stream
  float* const pout = out + pixBase * 3 + lane * 4;      // b128 store base

  float* const wlds = &lds[wave * 64];

#pragma unroll
  for (int t = 0; t < kTilesPerWave; ++t) {
    // Per-tile deltas: +32 dwords on both input streams, +48 floats on output.
    const int d = 32 * t;

    // Streaming prefetch ~2 tiles ahead on the strided Y/V stream
    // (global_prefetch_b8; speculative, OOB at frame end is dropped).
    __builtin_prefetch(p0 + d + 64, 0, 0);

    // Non-temporal streaming loads; lanes0-15 fetch Y, lanes16-31 fetch V.
    const float s0 = (float)__builtin_nontemporal_load(p0 + d);
    const float su = (float)__builtin_nontemporal_load(pu + d);

    v2f B;
    B.x = s0;                 // K0 = Y | K2 = V
    B.y = half ? 1.0f : su;   // K1 = U | K3 = 1

    v8f C = {};
    v8f D = __builtin_amdgcn_wmma_f32_16x16x4_f32(
        /*neg_a=*/false, A, /*neg_b=*/false, B,
        /*c_mod=*/(short)0, C, /*reuse_a=*/false, /*reuse_b=*/false);

    // Planar (V0=R,V1=G,V2=B across lanes 0-15) -> interleaved via LDS.
    // dword addresses 3*l+c are distinct mod 64 -> conflict-free.
    if (!half) {
      wlds[3 * l16 + 0] = D[0];
      wlds[3 * l16 + 1] = D[1];
      wlds[3 * l16 + 2] = D[2];
    }
    wave_lds_fence();
    if (lane < 12) {
      const v4f q = ((const v4f*)wlds)[lane];    // ds_load_b128, conflict-free
      __builtin_nontemporal_store(q, (v4f*)(pout + 48 * t));
    }
    wave_lds_fence();  // WAR guard before next tile's LDS writes
  }
}

extern "C" void kernel_launch(void* const* d_in, const int* in_sizes, int n_in,
                              void* d_out, int out_size, void* d_ws, size_t ws_size,
                              hipStream_t stream) {
  (void)in_sizes; (void)n_in; (void)out_size; (void)d_ws; (void)ws_size;
  const int* bytes = (const int*)d_in[0];
  const float* mat = (const float*)d_in[1];
  const float* off = (const float*)d_in[2];
  float* out = (float*)d_out;

  const int totalPix = kHalfH * kHalfW;          // 8,294,400
  const int blocks = totalPix / kPixPerBlock;    // 8100, exact cover
  nv12_to_rgb_wmma<<<blocks, kBlockThreads, 0, stream>>>(bytes, mat, off, out);
}